// MoE_Layer_32495722561822
// MI455X (gfx1250) — compile-verified
//
#include <hip/hip_runtime.h>
#include <hip/hip_bf16.h>

// N=65536, D=2048, E=8, TOP_K=2
#define MOE_N 65536
#define MOE_D 2048
#define MOE_E 8

typedef __attribute__((ext_vector_type(2))) float v2f;
typedef __attribute__((ext_vector_type(8))) float v8f;

// One wave computes a 16-token x 16-expert logits tile with
// V_WMMA_F32_16X16X4_F32 over K=2048, then softmax+top2 via wave32 shuffles.
// Block = 256 threads = 8 waves = 128 tokens. W (64KB) staged in LDS.
__global__ __launch_bounds__(256) void moe_router_wmma(
    const float* __restrict__ x,   // [N, D]
    const float* __restrict__ W,   // [E, D]
    const float* __restrict__ b,   // [E]
    float* __restrict__ out)       // [N*2 vals][N*2 idx][N*D x-copy]
{
    __shared__ float lds_w[MOE_E * MOE_D];   // 64 KB of 320 KB WGP LDS

    const int tid = threadIdx.x;

    // Stage W into LDS with 128-bit loads (8*2048*4B / 256 thr = 16 float4 each).
    {
        const float4* Wv = (const float4*)W;
        float4* Lv = (float4*)lds_w;
        #pragma unroll
        for (int i = 0; i < (MOE_E * MOE_D / 4) / 256; ++i)
            Lv[tid + i * 256] = Wv[tid + i * 256];
    }
    __syncthreads();

    const int wave    = tid >> 5;
    const int lane    = tid & 31;
    const int tileRow = (blockIdx.x * 8 + wave) * 16;

    const int m     = lane & 15;   // row within tile (A) / expert column (B)
    const int khalf = lane >> 4;   // 0: K={0,1}, 1: K={2,3} in the fragment

    // A: per-lane contiguous float2 walk along one token row.
    const float* xrow = x + (size_t)(tileRow + m) * MOE_D + 2 * khalf;
    // B: expert rows from LDS; columns 8..15 duplicate experts 0..7 (never read).
    const float* wrow = lds_w + (size_t)(m & 7) * MOE_D + 2 * khalf;

    v8f c0 = {0.f, 0.f, 0.f, 0.f, 0.f, 0.f, 0.f, 0.f};
    v8f c1 = {0.f, 0.f, 0.f, 0.f, 0.f, 0.f, 0.f, 0.f};

    // K loop: 512 WMMAs, two accumulators to break the D->C RAW chain.
    for (int k0 = 0; k0 < MOE_D; k0 += 8) {
        v2f a0, b0, a1, b1;
        a0.x = xrow[k0 + 0];     a0.y = xrow[k0 + 1];
        b0.x = wrow[k0 + 0];     b0.y = wrow[k0 + 1];
        a1.x = xrow[k0 + 4];     a1.y = xrow[k0 + 5];
        b1.x = wrow[k0 + 4];     b1.y = wrow[k0 + 5];
        c0 = __builtin_amdgcn_wmma_f32_16x16x4_f32(
                 false, a0, false, b0, (short)0, c0, false, false);
        c1 = __builtin_amdgcn_wmma_f32_16x16x4_f32(
                 false, a1, false, b1, (short)0, c1, false, false);
    }
    #pragma unroll
    for (int r = 0; r < 8; ++r) c0[r] += c1[r];

    // C/D layout: VGPR r, lanes 0-15 -> logits[row r][expert lane];
    //             lanes 16-31 -> logits[row 8+r][expert lane-16].
    const int  g      = lane & 7;                    // expert id in 8-lane group
    const bool writer = ((lane & 15) == 0);          // lanes 0 and 16
    const float bias  = b[g];

    #pragma unroll
    for (int r = 0; r < 8; ++r) {
        float l = c0[r] + bias;

        // softmax over 8 experts (8-lane xor reductions, wave32)
        float mx = l;
        #pragma unroll
        for (int off = 1; off < 8; off <<= 1)
            mx = fmaxf(mx, __shfl_xor(mx, off, 8));
        float e = __expf(l - mx);
        float s = e;
        #pragma unroll
        for (int off = 1; off < 8; off <<= 1)
            s += __shfl_xor(s, off, 8);
        float gate = e / s;

        // top-1 (tie-break: lower index, matching jax.lax.top_k)
        float v1 = gate; int i1 = g;
        #pragma unroll
        for (int off = 1; off < 8; off <<= 1) {
            float ov = __shfl_xor(v1, off, 8);
            int   oi = __shfl_xor(i1, off, 8);
            if (ov > v1 || (ov == v1 && oi < i1)) { v1 = ov; i1 = oi; }
        }
        // top-2
        float v2 = (g == i1) ? -1.0f : gate; int i2 = g;
        #pragma unroll
        for (int off = 1; off < 8; off <<= 1) {
            float ov = __shfl_xor(v2, off, 8);
            int   oi = __shfl_xor(i2, off, 8);
            if (ov > v2 || (ov == v2 && oi < i2)) { v2 = ov; i2 = oi; }
        }

        if (writer) {
            const int row = tileRow + r + khalf * 8;
            out[(size_t)row * 2 + 0] = v1;
            out[(size_t)row * 2 + 1] = v2;
            out[(size_t)2 * MOE_N + (size_t)row * 2 + 0] = (float)i1;
            out[(size_t)2 * MOE_N + (size_t)row * 2 + 1] = (float)i2;
        }
    }
}

// Passthrough: copy x (512 MB) -> output tail. This is the bandwidth roof.
__global__ __launch_bounds__(256) void moe_copy_x(
    const float4* __restrict__ src, float4* __restrict__ dst, size_t n4)
{
    size_t i = (size_t)blockIdx.x * 256 + threadIdx.x;
    if (i < n4) dst[i] = src[i];
}

extern "C" void kernel_launch(void* const* d_in, const int* in_sizes, int n_in,
                              void* d_out, int out_size, void* d_ws, size_t ws_size,
                              hipStream_t stream) {
    const float* x = (const float*)d_in[0];   // [N, D]
    const float* W = (const float*)d_in[1];   // [E, D]
    const float* b = (const float*)d_in[2];   // [E]
    float* out = (float*)d_out;

    // Router: 512 blocks x 256 threads -> 4096 waves, one 16-token tile each.
    moe_router_wmma<<<MOE_N / 128, 256, 0, stream>>>(x, W, b, out);

    // x passthrough as float4 stream.
    const size_t n4 = (size_t)MOE_N * MOE_D / 4;          // 33,554,432
    const int blocks = (int)((n4 + 255) / 256);           // 131,072
    moe_copy_x<<<blocks, 256, 0, stream>>>(
        (const float4*)x, (float4*)(out + (size_t)4 * MOE_N), n4);
}